// DeformableTransformer_33861522162330
// MI455X (gfx1250) — compile-verified
//
#include <hip/hip_runtime.h>
#include <math.h>

typedef float v2f __attribute__((ext_vector_type(2)));
typedef float v8f __attribute__((ext_vector_type(8)));

#define D_MODEL 256
#define N_HEADS 8
#define DH 32
#define D_FFN 1024
#define BATCH 8
#define LQ 1024
#define S_TOT 21760          // 128*128 + 64*64 + 32*32 + 16*16
#define BLQ (BATCH*LQ)       // 8192

// ---------------------------------------------------------------------------
// Register-blocked fp32 WMMA GEMM:  Y[M,N] = X[M,K] * op(W)^T + bias (+ReLU)
//   TRANSB==0 : B[k][n] = W[n*ldw + k]   (W is N x K row-major, "W^T" GEMM)
//   TRANSB==1 : B[k][n] = W[k*ldw + n]   (W is K x N row-major, plain X*W)
// One wave computes a 64x32 output block = 4(M) x 2(N) tiles of 16x16 via
// V_WMMA_F32_16X16X4_F32, stepping K by 4. Per K-step: 4 A-frag + 2 B-frag
// loads feed 8 WMMAs -> ~10.7 FLOP/byte at L1 (vs 4 for unblocked), and the
// 8 independent accumulators remove WMMA->WMMA RAW hazards.
// A-matrix per-lane layout (ISA 7.12.2, 32-bit A 16x4): lane<16 -> K {0,1},
// lane>=16 -> K {2,3} of row M=lane&15 => one b64 load per fragment.
// blockIdx.y applies (head) strides sX/sW/sY. Requires 64|M, 32|N, 4|K.
// ---------------------------------------------------------------------------
template<int TRANSB, int RELU>
__global__ __launch_bounds__(256)
void gemm64x32(const float* __restrict__ X, const float* __restrict__ W,
               const float* __restrict__ bias, float* __restrict__ Y,
               int M, int N, int K, int ldx, int ldw, int ldy,
               long long sX, long long sW, long long sY)
{
    X += (long long)blockIdx.y * sX;
    W += (long long)blockIdx.y * sW;
    Y += (long long)blockIdx.y * sY;

    const int blocksN = N >> 5;                 // 32-column blocks
    const int nBlocks = (M >> 6) * blocksN;     // 64-row blocks
    const int wave    = blockIdx.x * (blockDim.x >> 5) + (threadIdx.x >> 5);
    if (wave >= nBlocks) return;                // wave-uniform: EXEC all-1s
    const int bm   = wave / blocksN;
    const int bn   = wave - bm * blocksN;
    const int lane = threadIdx.x & 31;
    const int l15  = lane & 15;
    const int half = lane >> 4;                 // selects K {0,1} vs {2,3}

    const float* xr = X + (size_t)(bm * 64 + l15) * ldx + half * 2;

    v8f acc[4][2];
    #pragma unroll
    for (int i = 0; i < 4; ++i)
        #pragma unroll
        for (int j = 0; j < 2; ++j)
            acc[i][j] = (v8f){0.f,0.f,0.f,0.f,0.f,0.f,0.f,0.f};

    if (TRANSB == 0) {
        const float* wr = W + (size_t)(bn * 32 + l15) * ldw + half * 2;
        #pragma unroll 2
        for (int k = 0; k < K; k += 4) {
            v2f a[4], b[2];
            #pragma unroll
            for (int i = 0; i < 4; ++i)
                a[i] = *(const v2f*)(xr + (size_t)(i * 16) * ldx + k);
            #pragma unroll
            for (int j = 0; j < 2; ++j)
                b[j] = *(const v2f*)(wr + (size_t)(j * 16) * ldw + k);
            #pragma unroll
            for (int i = 0; i < 4; ++i)
                #pragma unroll
                for (int j = 0; j < 2; ++j)
                    acc[i][j] = __builtin_amdgcn_wmma_f32_16x16x4_f32(
                        false, a[i], false, b[j], (short)0, acc[i][j], false, false);
        }
    } else {
        const float* wp = W + (size_t)(half * 2) * ldw + bn * 32 + l15;
        #pragma unroll 2
        for (int k = 0; k < K; k += 4) {
            v2f a[4], b[2];
            #pragma unroll
            for (int i = 0; i < 4; ++i)
                a[i] = *(const v2f*)(xr + (size_t)(i * 16) * ldx + k);
            #pragma unroll
            for (int j = 0; j < 2; ++j) {
                b[j].x = wp[(size_t)k * ldw + j * 16];
                b[j].y = wp[(size_t)k * ldw + ldw + j * 16];
            }
            #pragma unroll
            for (int i = 0; i < 4; ++i)
                #pragma unroll
                for (int j = 0; j < 2; ++j)
                    acc[i][j] = __builtin_amdgcn_wmma_f32_16x16x4_f32(
                        false, a[i], false, b[j], (short)0, acc[i][j], false, false);
        }
    }

    #pragma unroll
    for (int j = 0; j < 2; ++j) {
        const int col = bn * 32 + j * 16 + l15;
        const float bv = bias ? bias[col] : 0.f;
        #pragma unroll
        for (int i = 0; i < 4; ++i) {
            float* yp = Y + (size_t)(bm * 64 + i * 16 + half * 8) * ldy + col;
            #pragma unroll
            for (int r = 0; r < 8; ++r) {       // C/D: VGPR r -> row half*8+r
                float v = acc[i][j][r] + bv;
                if (RELU) v = fmaxf(v, 0.f);
                yp[(size_t)r * ldy] = v;
            }
        }
    }
}

// ---------------------------------------------------------------------------
__global__ __launch_bounds__(256)
void add2(const float* __restrict__ a, const float* __restrict__ b,
          float* __restrict__ o, size_t n)
{
    size_t i = (size_t)blockIdx.x * blockDim.x + threadIdx.x;
    if (i < n) o[i] = a[i] + b[i];
}

// Row softmax over rowLen elements with pre-scale (used for attention scores).
__global__ __launch_bounds__(256)
void softmax_rows(float* __restrict__ x, int rowLen, float scale)
{
    float* p = x + (size_t)blockIdx.x * rowLen;
    __shared__ float red[8];
    const int t = threadIdx.x;

    float m = -3.402823466e38f;
    for (int i = t; i < rowLen; i += 256) m = fmaxf(m, p[i] * scale);
    for (int o = 16; o; o >>= 1) m = fmaxf(m, __shfl_xor(m, o, 32));
    if ((t & 31) == 0) red[t >> 5] = m;
    __syncthreads();
    float rm = red[0];
    #pragma unroll
    for (int i = 1; i < 8; ++i) rm = fmaxf(rm, red[i]);
    __syncthreads();

    float s = 0.f;
    for (int i = t; i < rowLen; i += 256) {
        float e = __expf(p[i] * scale - rm);
        p[i] = e;
        s += e;
    }
    for (int o = 16; o; o >>= 1) s += __shfl_xor(s, o, 32);
    if ((t & 31) == 0) red[t >> 5] = s;
    __syncthreads();
    float tot = 0.f;
    #pragma unroll
    for (int i = 0; i < 8; ++i) tot += red[i];
    const float inv = 1.f / tot;
    for (int i = t; i < rowLen; i += 256) p[i] *= inv;
}

// Fused residual-add + LayerNorm over d=256 (one row per 256-thread block).
__global__ __launch_bounds__(256)
void add_layernorm(const float* __restrict__ a, const float* __restrict__ r,
                   const float* __restrict__ g, const float* __restrict__ bt,
                   float* __restrict__ out)
{
    const size_t row = blockIdx.x;
    const int t = threadIdx.x;
    __shared__ float red[8];

    const float x = a[row * 256 + t] + r[row * 256 + t];
    float s = x;
    for (int o = 16; o; o >>= 1) s += __shfl_xor(s, o, 32);
    if ((t & 31) == 0) red[t >> 5] = s;
    __syncthreads();
    float tot = 0.f;
    #pragma unroll
    for (int i = 0; i < 8; ++i) tot += red[i];
    const float mu = tot * (1.f / 256.f);
    const float dx = x - mu;
    __syncthreads();

    s = dx * dx;
    for (int o = 16; o; o >>= 1) s += __shfl_xor(s, o, 32);
    if ((t & 31) == 0) red[t >> 5] = s;
    __syncthreads();
    tot = 0.f;
    #pragma unroll
    for (int i = 0; i < 8; ++i) tot += red[i];
    const float var = tot * (1.f / 256.f);
    out[row * 256 + t] = dx * rsqrtf(var + 1e-5f) * g[t] + bt[t];
}

// Softmax over the 16 contiguous (level,point) logits per (b,q,h).
__global__ __launch_bounds__(256)
void softmax16(float* __restrict__ x, int nRows)
{
    int r = blockIdx.x * blockDim.x + threadIdx.x;
    if (r >= nRows) return;
    float* p = x + (size_t)r * 16;
    float m = p[0];
    #pragma unroll
    for (int i = 1; i < 16; ++i) m = fmaxf(m, p[i]);
    float e[16], s = 0.f;
    #pragma unroll
    for (int i = 0; i < 16; ++i) { e[i] = __expf(p[i] - m); s += e[i]; }
    const float inv = 1.f / s;
    #pragma unroll
    for (int i = 0; i < 16; ++i) p[i] = e[i] * inv;
}

// Multi-scale deformable sampling. One thread per (b,q,h,c); the 32 lanes of
// a wave cover the 32 channels of one head -> corner gathers are 128B chunks.
__global__ __launch_bounds__(256)
void ms_deform(const float* __restrict__ val, const float* __restrict__ refp,
               const float* __restrict__ off, const float* __restrict__ aw,
               float* __restrict__ out)
{
    const size_t tid = (size_t)blockIdx.x * 256 + threadIdx.x;
    const int c  = (int)(tid & 31);
    const int h  = (int)((tid >> 5) & 7);
    const size_t bq = tid >> 8;                  // b*LQ + q
    const size_t b  = bq >> 10;

    const int HH[4] = {128, 64, 32, 16};
    const int WW[4] = {128, 64, 32, 16};
    const int ST[4] = {0, 16384, 20480, 21504};

    const float* refRow = refp + bq * 8;                 // (4 levels, 2)
    const float* offRow = off  + bq * 256 + h * 32;      // (4,4,2) per head
    const float* awRow  = aw   + bq * 128 + h * 16;      // (4,4)   per head
    float acc = 0.f;

    #pragma unroll
    for (int l = 0; l < 4; ++l) {
        const int Wi = WW[l], Hi = HH[l];
        const float rx = refRow[l * 2 + 0];
        const float ry = refRow[l * 2 + 1];
        const float* vb = val + ((size_t)b * S_TOT + ST[l]) * 256 + h * 32 + c;
        #pragma unroll
        for (int p = 0; p < 4; ++p) {
            // loc = ref + off/[W,H];  x = loc.x*W - 0.5 = ref.x*W + off.x - 0.5
            const float x = rx * (float)Wi + offRow[l * 8 + p * 2 + 0] - 0.5f;
            const float y = ry * (float)Hi + offRow[l * 8 + p * 2 + 1] - 0.5f;
            const float xf = floorf(x), yf = floorf(y);
            const float tx = x - xf, ty = y - yf;
            const int x0 = (int)xf, y0 = (int)yf;
            const bool vx0 = (x0 >= 0) && (x0 < Wi);
            const bool vx1 = (x0 + 1 >= 0) && (x0 + 1 < Wi);
            const bool vy0 = (y0 >= 0) && (y0 < Hi);
            const bool vy1 = (y0 + 1 >= 0) && (y0 + 1 < Hi);
            float s00 = 0.f, s10 = 0.f, s01 = 0.f, s11 = 0.f;
            if (vy0) {
                if (vx0) s00 = vb[(size_t)(y0 * Wi + x0) * 256];
                if (vx1) s10 = vb[(size_t)(y0 * Wi + x0 + 1) * 256];
            }
            if (vy1) {
                if (vx0) s01 = vb[(size_t)((y0 + 1) * Wi + x0) * 256];
                if (vx1) s11 = vb[(size_t)((y0 + 1) * Wi + x0 + 1) * 256];
            }
            const float bil = s00 * (1.f - tx) * (1.f - ty) + s10 * tx * (1.f - ty)
                            + s01 * (1.f - tx) * ty         + s11 * tx * ty;
            acc += awRow[l * 4 + p] * bil;
        }
    }
    out[bq * 256 + h * 32 + c] = acc;
}

// ---------------------------------------------------------------------------
extern "C" void kernel_launch(void* const* d_in, const int* in_sizes, int n_in,
                              void* d_out, int out_size, void* d_ws, size_t ws_size,
                              hipStream_t stream)
{
    (void)in_sizes; (void)n_in; (void)out_size; (void)ws_size;
    const float* tgt    = (const float*)d_in[0];
    const float* qpos   = (const float*)d_in[1];
    const float* refp   = (const float*)d_in[2];
    const float* src    = (const float*)d_in[3];
    const float* w_in   = (const float*)d_in[4];
    const float* b_in   = (const float*)d_in[5];
    const float* w_out  = (const float*)d_in[6];
    const float* b_out  = (const float*)d_in[7];
    const float* ln1g   = (const float*)d_in[8];
    const float* ln1b   = (const float*)d_in[9];
    const float* ln2g   = (const float*)d_in[10];
    const float* ln2b   = (const float*)d_in[11];
    const float* ln3g   = (const float*)d_in[12];
    const float* ln3b   = (const float*)d_in[13];
    const float* w_val  = (const float*)d_in[14];
    const float* b_val  = (const float*)d_in[15];
    const float* w_off  = (const float*)d_in[16];
    const float* b_off  = (const float*)d_in[17];
    const float* w_attw = (const float*)d_in[18];
    const float* b_attw = (const float*)d_in[19];
    const float* w_oc   = (const float*)d_in[20];
    const float* b_oc   = (const float*)d_in[21];
    const float* w1     = (const float*)d_in[22];
    const float* b1     = (const float*)d_in[23];
    const float* w2     = (const float*)d_in[24];
    const float* b2     = (const float*)d_in[25];

    float* ws = (float*)d_ws;
    size_t o = 0;
    float* QK   = ws + o; o += (size_t)BLQ * 256;            // tgt + query_pos
    float* QKP  = ws + o; o += (size_t)BLQ * 512;            // [q | k] projection
    float* VB   = ws + o; o += (size_t)BLQ * 256;            // v projection
    float* SC   = ws + o; o += (size_t)N_HEADS * LQ * LQ;    // per-batch scores
    float* OB   = ws + o; o += (size_t)BLQ * 256;            // attention output
    float* T2   = ws + o; o += (size_t)BLQ * 256;            // residual branch
    float* TGT2 = ws + o; o += (size_t)BLQ * 256;            // after ln2
    float* QRY  = ws + o; o += (size_t)BLQ * 256;            // tgt2 + query_pos
    float* VAL  = ws + o; o += (size_t)BATCH * S_TOT * 256;  // value projection
    float* OFFB = ws + o; o += (size_t)BLQ * 256;            // sampling offsets
    float* AWB  = ws + o; o += (size_t)BLQ * 128;            // attn weights
    float* DFO  = ws + o; o += (size_t)BLQ * 256;            // deform output
    float* TGT3 = ws + o; o += (size_t)BLQ * 256;            // after ln1
    float* FFN  = ws + o; o += (size_t)BLQ * 1024;           // ffn hidden
    float* T2B  = ws + o; o += (size_t)BLQ * 256;            // ffn out
    (void)o;

    // grid for the 64x32-blocked GEMM (8 waves per 256-thread block)
    auto G = [](int M, int N) { return dim3((unsigned)(((M / 64) * (N / 32) + 7) / 8)); };

    // 1. qk = tgt + query_pos
    add2<<<dim3(BLQ), 256, 0, stream>>>(tgt, qpos, QK, (size_t)BLQ * 256);
    // 2. [q|k] = qk @ w_in[0:512].T + b_in[0:512]
    gemm64x32<0,0><<<G(BLQ,512),256,0,stream>>>(QK, w_in, b_in, QKP,
        BLQ,512,256, 256,256,512, 0,0,0);
    // 3. v = tgt @ w_in[512:768].T + b_in[512:]
    gemm64x32<0,0><<<G(BLQ,256),256,0,stream>>>(tgt, w_in + 512*256, b_in + 512, VB,
        BLQ,256,256, 256,256,256, 0,0,0);

    // 4. attention, chunked per batch; gridDim.y = heads via stride args
    const float scale = 0.17677669529663687f;  // 1/sqrt(32)
    for (int b = 0; b < BATCH; ++b) {
        const float* qb = QKP + (size_t)b * LQ * 512;
        const float* kb = qb + 256;
        const float* vb = VB + (size_t)b * LQ * 256;
        float*       ob = OB + (size_t)b * LQ * 256;
        dim3 gs((unsigned)(((LQ/64)*(LQ/32) + 7) / 8), N_HEADS);
        gemm64x32<0,0><<<gs,256,0,stream>>>(qb, kb, nullptr, SC,
            LQ,LQ,DH, 512,512,LQ, 32,32,(long long)LQ*LQ);
        softmax_rows<<<dim3(N_HEADS*LQ),256,0,stream>>>(SC, LQ, scale);
        dim3 go((unsigned)(((LQ/64)*(DH/32) + 7) / 8), N_HEADS);
        gemm64x32<1,0><<<go,256,0,stream>>>(SC, vb, nullptr, ob,
            LQ,DH,LQ, LQ,256,256, (long long)LQ*LQ,32,32);
    }

    // 5. t2 = o @ w_out.T + b_out
    gemm64x32<0,0><<<G(BLQ,256),256,0,stream>>>(OB, w_out, b_out, T2,
        BLQ,256,256, 256,256,256, 0,0,0);
    // 6. tgt2 = LN(tgt + t2; ln2)
    add_layernorm<<<dim3(BLQ),256,0,stream>>>(tgt, T2, ln2g, ln2b, TGT2);
    // 7. query = tgt2 + query_pos
    add2<<<dim3(BLQ),256,0,stream>>>(TGT2, qpos, QRY, (size_t)BLQ * 256);
    // 8. value = src @ w_val.T + b_val   (largest GEMM: 22.8 GFLOP)
    gemm64x32<0,0><<<G(BATCH*S_TOT,256),256,0,stream>>>(src, w_val, b_val, VAL,
        BATCH*S_TOT,256,256, 256,256,256, 0,0,0);
    // 9. off = query @ w_off.T + b_off
    gemm64x32<0,0><<<G(BLQ,256),256,0,stream>>>(QRY, w_off, b_off, OFFB,
        BLQ,256,256, 256,256,256, 0,0,0);
    // 10. aw logits = query @ w_attw.T + b_attw, then softmax over 16
    gemm64x32<0,0><<<G(BLQ,128),256,0,stream>>>(QRY, w_attw, b_attw, AWB,
        BLQ,128,256, 256,256,128, 0,0,0);
    softmax16<<<dim3((BLQ*N_HEADS + 255)/256),256,0,stream>>>(AWB, BLQ*N_HEADS);
    // 11. deformable sampling
    ms_deform<<<dim3(BLQ),256,0,stream>>>(VAL, refp, OFFB, AWB, DFO);
    // 12. ca = dfo @ w_oc.T + b_oc   (reuse T2)
    gemm64x32<0,0><<<G(BLQ,256),256,0,stream>>>(DFO, w_oc, b_oc, T2,
        BLQ,256,256, 256,256,256, 0,0,0);
    // 13. tgt3 = LN(tgt2 + ca; ln1)
    add_layernorm<<<dim3(BLQ),256,0,stream>>>(TGT2, T2, ln1g, ln1b, TGT3);
    // 14. FFN: h = relu(tgt3 @ w1.T + b1); t2 = h @ w2.T + b2
    gemm64x32<0,1><<<G(BLQ,1024),256,0,stream>>>(TGT3, w1, b1, FFN,
        BLQ,1024,256, 256,256,1024, 0,0,0);
    gemm64x32<0,0><<<G(BLQ,256),256,0,stream>>>(FFN, w2, b2, T2B,
        BLQ,256,1024, 1024,1024,256, 0,0,0);
    // 15. out = LN(tgt3 + t2; ln3)
    add_layernorm<<<dim3(BLQ),256,0,stream>>>(TGT3, T2B, ln3g, ln3b, (float*)d_out);
}